// SpatioConvLayer_88184268521809
// MI455X (gfx1250) — compile-verified
//
#include <hip/hip_runtime.h>
#include <hip/hip_bf16.h>

typedef __attribute__((ext_vector_type(2))) float v2f;
typedef __attribute__((ext_vector_type(8))) float v8f;

static constexpr int BQ = 4;        // batch
static constexpr int NN = 10000;    // nodes
static constexpr int CC = 128;      // channels (c_in == c_out)
static constexpr int EE = 320000;   // edges
static constexpr int FF = 16;       // edge feature size

// ---------------------------------------------------------------------------
// 1) zero the degree array
__global__ void zero_deg_kernel(float* __restrict__ deg) {
    int i = blockIdx.x * blockDim.x + threadIdx.x;
    if (i < NN) deg[i] = 0.0f;
}

// 2) deg[src] += 1  (source-node out-degree, matching reference)
__global__ void count_deg_kernel(const long long* __restrict__ ei,
                                 float* __restrict__ deg) {
    int e = blockIdx.x * blockDim.x + threadIdx.x;
    if (e < EE) atomicAdd(&deg[(int)ei[e]], 1.0f);
}

// 3) per-edge scalar weight: (edge_attr . We + be) * sqrt(deg[src])*sqrt(deg[dst])
__global__ void edge_weight_kernel(const float* __restrict__ ea,
                                   const float* __restrict__ We,
                                   const float* __restrict__ be,
                                   const long long* __restrict__ ei,
                                   const float* __restrict__ deg,
                                   float* __restrict__ w) {
    int e = blockIdx.x * blockDim.x + threadIdx.x;
    if (e >= EE) return;
    float acc = be[0];
    const float* row = ea + (size_t)e * FF;
#pragma unroll
    for (int i = 0; i < FF; ++i) acc += row[i] * We[i];
    int s = (int)ei[e];
    int d = (int)ei[EE + e];
    float nrm = sqrtf(deg[s]) * sqrtf(deg[d]);
    w[e] = acc * nrm;
}

// 4) out[b,n,c] = bias[c]   (scatter accumulates on top, relu at the end)
__global__ void init_out_kernel(float* __restrict__ out,
                                const float* __restrict__ bias) {
    int i = blockIdx.x * blockDim.x + threadIdx.x;     // float4 index
    const int total4 = BQ * NN * CC / 4;
    if (i >= total4) return;
    int c = (i * 4) & (CC - 1);
    float4 bv = *reinterpret_cast<const float4*>(bias + c);
    reinterpret_cast<float4*>(out)[i] = bv;
}

// 5) xl = x @ W via fp32 WMMA (16x16 tile per wave, K-loop of 4)
__global__ __launch_bounds__(256)
void gemm_xw_wmma_kernel(const float* __restrict__ X,
                         const float* __restrict__ W,
                         float* __restrict__ XL) {
    const int lane  = threadIdx.x & 31;
    const int gwave = blockIdx.x * 8 + (threadIdx.x >> 5);
    const int tm = gwave >> 3;                  // 40000/16 = 2500 row tiles
    const int tn = gwave & 7;                   // 128/16   = 8 col tiles
    const int M  = BQ * NN;
    if (tm * 16 >= M) return;                   // wave-uniform -> EXEC all-1s

    const int half = lane >> 4;                 // 0: K=0,1 ; 1: K=2,3
    const int l15  = lane & 15;
    const int row  = tm * 16 + l15;             // A: lane = M row
    const int col  = tn * 16 + l15;             // B: lane = N col
    const int kh   = half * 2;
    const float* xrow = X + (size_t)row * CC;

    v8f c = {};
#pragma unroll 4
    for (int k = 0; k < CC; k += 4) {
        v2f a, b;
        a[0] = xrow[k + kh];
        a[1] = xrow[k + kh + 1];
        b[0] = W[(size_t)(k + kh)     * CC + col];
        b[1] = W[(size_t)(k + kh + 1) * CC + col];
        // 8 args: (neg_a, A, neg_b, B, c_mod, C, reuse_a, reuse_b)
        c = __builtin_amdgcn_wmma_f32_16x16x4_f32(false, a, false, b,
                                                  (short)0, c, false, false);
    }
    // C/D layout: VGPR v -> row v (lanes 0-15), row v+8 (lanes 16-31)
#pragma unroll
    for (int v = 0; v < 8; ++v) {
        int r = tm * 16 + v + half * 8;
        XL[(size_t)r * CC + col] = c[v];
    }
}

// 6) scatter: one wave32 per edge, lane owns 4 contiguous channels, loop batches.
//    Gather float4 from xl (L2-resident), 4 coalesced f32 atomics into out.
__global__ __launch_bounds__(256)
void scatter_kernel(const float* __restrict__ xl,
                    const float* __restrict__ w,
                    const long long* __restrict__ ei,
                    float* __restrict__ out) {
    const int e = (int)((blockIdx.x * (size_t)blockDim.x + threadIdx.x) >> 5);
    const int lane = threadIdx.x & 31;
    if (e >= EE) return;
    const int src = (int)ei[e];
    const int dst = (int)ei[EE + e];
    const float we = w[e];
    const int c = lane * 4;
#pragma unroll
    for (int b = 0; b < BQ; ++b) {
        const float4 v = *reinterpret_cast<const float4*>(
            xl + ((size_t)b * NN + src) * CC + c);
        float* o = out + ((size_t)b * NN + dst) * CC + c;
        atomicAdd(o + 0, v.x * we);
        atomicAdd(o + 1, v.y * we);
        atomicAdd(o + 2, v.z * we);
        atomicAdd(o + 3, v.w * we);
    }
}

// 7) in-place ReLU
__global__ void relu_kernel(float* __restrict__ out) {
    int i = blockIdx.x * blockDim.x + threadIdx.x;     // float4 index
    const int total4 = BQ * NN * CC / 4;
    if (i >= total4) return;
    float4 v = reinterpret_cast<float4*>(out)[i];
    v.x = fmaxf(v.x, 0.0f); v.y = fmaxf(v.y, 0.0f);
    v.z = fmaxf(v.z, 0.0f); v.w = fmaxf(v.w, 0.0f);
    reinterpret_cast<float4*>(out)[i] = v;
}

extern "C" void kernel_launch(void* const* d_in, const int* in_sizes, int n_in,
                              void* d_out, int out_size, void* d_ws, size_t ws_size,
                              hipStream_t stream) {
    // setup_inputs order: x, edge_index, edge_attr, W, b, We, be
    const float*     x   = (const float*)d_in[0];        // [B,N,128] f32
    const long long* ei  = (const long long*)d_in[1];    // [2,E] i64
    const float*     ea  = (const float*)d_in[2];        // [E,16] f32
    const float*     W   = (const float*)d_in[3];        // [128,128]
    const float*     bia = (const float*)d_in[4];        // [128]
    const float*     We  = (const float*)d_in[5];        // [16]
    const float*     be  = (const float*)d_in[6];        // [1]
    float*           out = (float*)d_out;                // [B,N,128]

    // workspace layout (floats): deg[N] | w[E] | xl[B*N*128]  (~21.8 MB)
    float* deg = (float*)d_ws;
    float* wE  = deg + NN;
    float* xl  = wE + EE;

    const int threads = 256;
    const int total4  = BQ * NN * CC / 4;

    zero_deg_kernel<<<(NN + threads - 1) / threads, threads, 0, stream>>>(deg);
    count_deg_kernel<<<(EE + threads - 1) / threads, threads, 0, stream>>>(ei, deg);
    edge_weight_kernel<<<(EE + threads - 1) / threads, threads, 0, stream>>>(
        ea, We, be, ei, deg, wE);
    init_out_kernel<<<(total4 + threads - 1) / threads, threads, 0, stream>>>(out, bia);

    // GEMM: 2500*8 = 20000 tiles, 8 waves/block -> 2500 blocks
    const int tiles = (BQ * NN / 16) * (CC / 16);
    gemm_xw_wmma_kernel<<<(tiles + 7) / 8, threads, 0, stream>>>(x, W, xl);

    // scatter: one wave per edge -> E*32 threads
    scatter_kernel<<<(int)(((size_t)EE * 32 + threads - 1) / threads), threads, 0,
                     stream>>>(xl, wE, ei, out);

    relu_kernel<<<(total4 + threads - 1) / threads, threads, 0, stream>>>(out);
}